// CrossAttentionTransformer_41686952575256
// MI455X (gfx1250) — compile-verified
//
#include <hip/hip_runtime.h>
#include <hip/hip_bf16.h>

// ---------------------------------------------------------------------------
// CrossAttentionTransformer forward for MI455X (gfx1250, wave32).
// fp32 WMMA (v_wmma_f32_16x16x4_f32) for all GEMMs: workload is HBM-bound
// (~2GB traffic vs ~90 GFLOP), so full-precision matrix ops are free.
// GEMM dims are template constants so all strides fold into immediates /
// constant-stride pointer increments (no 64-bit address math in the K loop).
// ---------------------------------------------------------------------------

typedef __attribute__((ext_vector_type(2))) float v2f;
typedef __attribute__((ext_vector_type(8))) float v8f;

#define LN_EPS 1e-5f
#define ATTN_SCALE 0.08838834764831845f  // 128^-0.5

__device__ __forceinline__ float wave_sum32(float v) {
#pragma unroll
  for (int off = 16; off > 0; off >>= 1) v += __shfl_xor(v, off, 32);
  return v;
}

// ---------------- LayerNorm: one wave per row of 128 -----------------------
__global__ void ln_kernel(const float* __restrict__ x, const float* __restrict__ g,
                          const float* __restrict__ b, float* __restrict__ y, int rows) {
  int w = (blockIdx.x * blockDim.x + threadIdx.x) >> 5;
  int lane = threadIdx.x & 31;
  if (w >= rows) return;
  float4 xv = *(const float4*)(x + (size_t)w * 128 + lane * 4);
  float mu = wave_sum32(xv.x + xv.y + xv.z + xv.w) * (1.0f / 128.0f);
  float dx = xv.x - mu, dy = xv.y - mu, dz = xv.z - mu, dw = xv.w - mu;
  float var = wave_sum32(dx * dx + dy * dy + dz * dz + dw * dw) * (1.0f / 128.0f);
  float rstd = rsqrtf(var + LN_EPS);
  float4 gv = *(const float4*)(g + lane * 4);
  float4 bv = *(const float4*)(b + lane * 4);
  float4 o;
  o.x = dx * rstd * gv.x + bv.x;
  o.y = dy * rstd * gv.y + bv.y;
  o.z = dz * rstd * gv.z + bv.z;
  o.w = dw * rstd * gv.w + bv.w;
  *(float4*)(y + (size_t)w * 128 + lane * 4) = o;
}

// ---------------- GEMM: Y = act(X[rows,DIN] @ W[DIN,DOUT] + bias) (+r1)(+r2)
// One wave -> one 16x16 output tile via V_WMMA_F32_16X16X4_F32.
// Block = 256 threads = 8 waves covering 16 rows x 128 cols.
// A layout: lane%16 = M row; VGPR {0,1} = K {0,1} (lanes<16) / K {2,3} (lanes>=16)
// B layout (symmetric by column): lane%16 = N col; same K split.
// C/D layout: lane%16 = N; acc[i] = row (i + 8*(lane>=16)).
template <int DIN, int DOUT, bool GELU>
__global__ void gemm_kernel(const float* __restrict__ X, const float* __restrict__ W,
                            const float* __restrict__ bias,
                            const float* __restrict__ r1, const float* __restrict__ r2,
                            float* __restrict__ Y, int rows) {
  int waveId = threadIdx.x >> 5;
  int lane = threadIdx.x & 31;
  int l16 = lane & 15;
  int half = lane >> 4;
  int row_base = blockIdx.x * 16;
  int col = blockIdx.y * 128 + waveId * 16 + l16;

  int rowA = row_base + l16;
  if (rowA >= rows) rowA = rows - 1;  // clamped load, stores are guarded
  const float* xp = X + (size_t)rowA * DIN + 2 * half;
  const float* wp = W + (2 * half) * DOUT + col;

  v8f acc = {};
#pragma unroll 8
  for (int k = 0; k < DIN; k += 4) {
    v2f a = *(const v2f*)(xp + k);  // X[rowA, k+2h], X[rowA, k+2h+1]
    v2f bmat;
    bmat.x = wp[k * DOUT];          // W[k+2h,   col]  (compile-time stride)
    bmat.y = wp[k * DOUT + DOUT];   // W[k+2h+1, col]
    acc = __builtin_amdgcn_wmma_f32_16x16x4_f32(false, a, false, bmat,
                                                (short)0, acc, false, false);
  }

  float bb = bias[col];
#pragma unroll
  for (int i = 0; i < 8; ++i) {
    int r = row_base + half * 8 + i;
    if (r < rows) {
      float val = acc[i] + bb;
      if (GELU) val = 0.5f * val * (1.0f + erff(val * 0.70710678118654752f));
      size_t idx = (size_t)r * DOUT + col;
      if (r1) val += r1[idx];
      if (r2) val += r2[idx];
      Y[idx] = val;
    }
  }
}

// ---------------- segment attention helpers --------------------------------
__device__ __forceinline__ unsigned enc_f32(float f) {
  unsigned b = __float_as_uint(f);
  return (b & 0x80000000u) ? ~b : (b | 0x80000000u);
}
__device__ __forceinline__ float dec_f32(unsigned u) {
  unsigned b = (u & 0x80000000u) ? (u ^ 0x80000000u) : ~u;
  return __uint_as_float(b);
}

// wave per edge: logit = scale * dot(Q[dst], K[src]); atomicMax segment max
__global__ void edge_logit_kernel(const int* __restrict__ src, const int* __restrict__ dst,
                                  const float* __restrict__ Q, const float* __restrict__ K,
                                  float* __restrict__ logit, unsigned* __restrict__ mseg,
                                  int E) {
  int e = (blockIdx.x * blockDim.x + threadIdx.x) >> 5;
  int lane = threadIdx.x & 31;
  if (e >= E) return;
  int s = src[e], d = dst[e];
  float4 q = *(const float4*)(Q + (size_t)d * 128 + lane * 4);
  float4 k = *(const float4*)(K + (size_t)s * 128 + lane * 4);
  float dot = wave_sum32(q.x * k.x + q.y * k.y + q.z * k.z + q.w * k.w) * ATTN_SCALE;
  if (lane == 0) {
    logit[e] = dot;
    atomicMax(mseg + d, enc_f32(dot));
  }
}

// thread per edge: ex = exp(logit - m[dst]); den[dst] += ex (overwrite logit)
__global__ void edge_exp_kernel(const int* __restrict__ dst, float* __restrict__ logit,
                                const unsigned* __restrict__ mseg, float* __restrict__ den,
                                int E) {
  int e = blockIdx.x * blockDim.x + threadIdx.x;
  if (e >= E) return;
  int d = dst[e];
  float ex = expf(logit[e] - dec_f32(mseg[d]));
  logit[e] = ex;
  atomicAdd(den + d, ex);
}

// wave per edge: out[dst] += (ex/den[dst]) * V[src]
__global__ void edge_scatter_kernel(const int* __restrict__ src, const int* __restrict__ dst,
                                    const float* __restrict__ ex, const float* __restrict__ den,
                                    const float* __restrict__ V, float* __restrict__ out, int E) {
  int e = (blockIdx.x * blockDim.x + threadIdx.x) >> 5;
  int lane = threadIdx.x & 31;
  if (e >= E) return;
  int s = src[e], d = dst[e];
  float alpha = ex[e] / (den[d] + 1e-16f);
  float4 v = *(const float4*)(V + (size_t)s * 128 + lane * 4);
  float* op = out + (size_t)d * 128 + lane * 4;
  atomicAdd(op + 0, alpha * v.x);
  atomicAdd(op + 1, alpha * v.y);
  atomicAdd(op + 2, alpha * v.z);
  atomicAdd(op + 3, alpha * v.w);
}

// ---------------- utility ---------------------------------------------------
__global__ void zero_kernel(float* __restrict__ p, long long n) {
  long long i = (long long)blockIdx.x * blockDim.x + threadIdx.x;
  if (i < n) p[i] = 0.0f;
}
__global__ void copy_kernel(const float* __restrict__ a, float* __restrict__ y, long long n) {
  long long i = (long long)blockIdx.x * blockDim.x + threadIdx.x;
  if (i < n) y[i] = a[i];
}

// ---------------- host orchestration ----------------------------------------
static inline void gemm_dd(const float* X, const float* W, const float* b,
                           const float* r1, const float* r2, float* Y,
                           int rows, hipStream_t s) {  // 128 -> 128
  gemm_kernel<128, 128, false><<<dim3((rows + 15) / 16, 1), 256, 0, s>>>(X, W, b, r1, r2, Y, rows);
}
static inline void gemm_up_gelu(const float* X, const float* W, const float* b,
                                float* Y, int rows, hipStream_t s) {  // 128 -> 512, GELU
  gemm_kernel<128, 512, true><<<dim3((rows + 15) / 16, 4), 256, 0, s>>>(X, W, b, nullptr, nullptr, Y, rows);
}
static inline void gemm_down(const float* X, const float* W, const float* b,
                             const float* r1, float* Y, int rows, hipStream_t s) {  // 512 -> 128
  gemm_kernel<512, 128, false><<<dim3((rows + 15) / 16, 1), 256, 0, s>>>(X, W, b, r1, nullptr, Y, rows);
}

static inline void seg_attn(const int* src, const int* dst, int E, int num_dst,
                            const float* Q, const float* K, const float* V,
                            float* out /*pre-initialized*/, float* elog,
                            unsigned* mseg, float* den, hipStream_t s) {
  int zg = (num_dst + 255) / 256;
  zero_kernel<<<zg, 256, 0, s>>>((float*)mseg, num_dst);  // enc(-maxfloat) > 0, so 0 == "-inf"
  zero_kernel<<<zg, 256, 0, s>>>(den, num_dst);
  int wg = (E + 7) / 8;  // 8 waves / block
  edge_logit_kernel<<<wg, 256, 0, s>>>(src, dst, Q, K, elog, mseg, E);
  edge_exp_kernel<<<(E + 255) / 256, 256, 0, s>>>(dst, elog, mseg, den, E);
  edge_scatter_kernel<<<wg, 256, 0, s>>>(src, dst, elog, den, V, out, E);
}

extern "C" void kernel_launch(void* const* d_in, const int* in_sizes, int n_in,
                              void* d_out, int out_size, void* d_ws, size_t ws_size,
                              hipStream_t stream) {
  // ---- inputs (T = 1) ----
  const float* node = (const float*)d_in[0];
  const float* tri = (const float*)d_in[1];
  const int N = in_sizes[0] / 128;
  const int M = in_sizes[1] / 128;
  const int* node_ei = (const int*)d_in[2]; const int E_NODE = in_sizes[2] / 2;
  const int* tri_ei  = (const int*)d_in[3]; const int E_TRI  = in_sizes[3] / 2;
  const int* nt_ei   = (const int*)d_in[4]; const int E_NT   = in_sizes[4] / 2;
  const int* tn_ei   = (const int*)d_in[5]; const int E_TN   = in_sizes[5] / 2;

  // ---- params: JAX pytree (sorted dict keys) starting at index 6 ----
  auto P = [&](int i) { return (const float*)d_in[i]; };
  const float *lnc_n_b = P(6),  *lnc_n_g = P(7);    // ln_node_cross
  const float *lnm_n_b = P(8),  *lnm_n_g = P(9);    // ln_node_mlp
  const float *lns_n_b = P(10), *lns_n_g = P(11);   // ln_node_self
  const float *lnc_t_b = P(12), *lnc_t_g = P(13);   // ln_tri_cross
  const float *lnm_t_b = P(14), *lnm_t_g = P(15);   // ln_tri_mlp
  const float *lns_t_b = P(16), *lns_t_g = P(17);   // ln_tri_self
  const float *mn_W1 = P(18), *mn_W2 = P(19), *mn_b1 = P(20), *mn_b2 = P(21);
  const float *mt_W1 = P(22), *mt_W2 = P(23), *mt_b1 = P(24), *mt_b2 = P(25);
  const float *n2t_Wk = P(26), *n2t_Wo = P(27), *n2t_Wq = P(28), *n2t_Wv = P(29);
  const float *n2t_bk = P(30), *n2t_bo = P(31), *n2t_bq = P(32), *n2t_bv = P(33);
  const float *na_Wk = P(34), *na_Wq = P(35), *na_Wv = P(36);
  const float *na_bk = P(37), *na_bq = P(38), *na_bv = P(39);
  const float *t2n_Wk = P(40), *t2n_Wo = P(41), *t2n_Wq = P(42), *t2n_Wv = P(43);
  const float *t2n_bk = P(44), *t2n_bo = P(45), *t2n_bq = P(46), *t2n_bv = P(47);
  const float *ta_Wk = P(48), *ta_Wq = P(49), *ta_Wv = P(50);
  const float *ta_bk = P(51), *ta_bq = P(52), *ta_bv = P(53);

  // ---- workspace layout ----
  const size_t NB = (size_t)N * 128, MB = (size_t)M * 128;
  const int NMmax = (M > N) ? M : N;
  int Emax = E_NODE;
  if (E_TRI > Emax) Emax = E_TRI;
  if (E_NT > Emax) Emax = E_NT;
  if (E_TN > Emax) Emax = E_TN;

  float* ws = (float*)d_ws;
  size_t off = 0;
  auto alloc = [&](size_t n) { float* p = ws + off; off += n; return p; };
  float* n_norm = alloc(NB);
  float* t_norm = alloc(MB);
  float* nQ = alloc(NB); float* nK = alloc(NB); float* nV = alloc(NB); float* nO = alloc(NB);
  float* mQ = alloc(MB); float* mK = alloc(MB); float* mV = alloc(MB); float* mO = alloc(MB);
  float* tmpN = alloc(NB);
  float* tmpM = alloc(MB);
  float* aggN = alloc(NB);
  float* aggM = alloc(MB);
  float* hid = alloc((size_t)NMmax * 512);
  float* elog = alloc((size_t)Emax);
  unsigned* mseg = (unsigned*)alloc((size_t)NMmax);
  float* den = alloc((size_t)NMmax);
  (void)ws_size; (void)n_in; (void)out_size;

  float* out_node = (float*)d_out;
  float* out_tri = out_node + NB;
  hipStream_t s = stream;
  const int LNB = 256;  // 8 rows per block for LN

  // ===== Phase A: pre-norms =====
  ln_kernel<<<(N + 7) / 8, LNB, 0, s>>>(node, lns_n_g, lns_n_b, n_norm, N);
  ln_kernel<<<(M + 7) / 8, LNB, 0, s>>>(tri, lns_t_g, lns_t_b, t_norm, M);

  // ===== Phase B: node self-attention (n1 = node + attn) =====
  gemm_dd(n_norm, na_Wq, na_bq, nullptr, nullptr, nQ, N, s);
  gemm_dd(n_norm, na_Wk, na_bk, nullptr, nullptr, nK, N, s);
  gemm_dd(n_norm, na_Wv, na_bv, nullptr, nullptr, nV, N, s);
  copy_kernel<<<(int)((NB + 255) / 256), 256, 0, s>>>(node, nO, (long long)NB);  // residual seed
  seg_attn(node_ei, node_ei + E_NODE, E_NODE, N, nQ, nK, nV, nO, elog, mseg, den, s);
  // nO == n1

  // ===== Phase C: triangle self-attention (tri1 = tri + attn) =====
  gemm_dd(t_norm, ta_Wq, ta_bq, nullptr, nullptr, mQ, M, s);
  gemm_dd(t_norm, ta_Wk, ta_bk, nullptr, nullptr, mK, M, s);
  gemm_dd(t_norm, ta_Wv, ta_bv, nullptr, nullptr, mV, M, s);
  copy_kernel<<<(int)((MB + 255) / 256), 256, 0, s>>>(tri, mO, (long long)MB);
  seg_attn(tri_ei, tri_ei + E_TRI, E_TRI, M, mQ, mK, mV, mO, elog, mseg, den, s);
  // mO == tri1

  // ===== Phase D: node->tri cross attention (dst = triangles) =====
  ln_kernel<<<(M + 7) / 8, LNB, 0, s>>>(mO, lnc_t_g, lnc_t_b, tmpM, M);  // q input
  gemm_dd(tmpM, n2t_Wq, n2t_bq, nullptr, nullptr, mQ, M, s);
  gemm_dd(n_norm, n2t_Wk, n2t_bk, nullptr, nullptr, nK, N, s);
  gemm_dd(n_norm, n2t_Wv, n2t_bv, nullptr, nullptr, nV, N, s);
  zero_kernel<<<(int)((MB + 255) / 256), 256, 0, s>>>(aggM, (long long)MB);
  seg_attn(nt_ei, nt_ei + E_NT, E_NT, M, mQ, nK, nV, aggM, elog, mseg, den, s);
  // tri2 = tri1 + (agg@Wo + bo + LN(tri1)); in-place into mO
  gemm_dd(aggM, n2t_Wo, n2t_bo, tmpM, mO, mO, M, s);

  // ===== Phase E: tri->node cross attention (dst = nodes) =====
  ln_kernel<<<(N + 7) / 8, LNB, 0, s>>>(nO, lnc_n_g, lnc_n_b, tmpN, N);
  gemm_dd(tmpN, t2n_Wq, t2n_bq, nullptr, nullptr, nQ, N, s);
  gemm_dd(t_norm, t2n_Wk, t2n_bk, nullptr, nullptr, mK, M, s);
  gemm_dd(t_norm, t2n_Wv, t2n_bv, nullptr, nullptr, mV, M, s);
  zero_kernel<<<(int)((NB + 255) / 256), 256, 0, s>>>(aggN, (long long)NB);
  seg_attn(tn_ei, tn_ei + E_TN, E_TN, N, nQ, mK, mV, aggN, elog, mseg, den, s);
  gemm_dd(aggN, t2n_Wo, t2n_bo, tmpN, nO, nO, N, s);  // n2 in-place

  // ===== Phase F: MLPs with pre-norm + residual to d_out =====
  ln_kernel<<<(N + 7) / 8, LNB, 0, s>>>(nO, lnm_n_g, lnm_n_b, tmpN, N);
  gemm_up_gelu(tmpN, mn_W1, mn_b1, hid, N, s);
  gemm_down(hid, mn_W2, mn_b2, nO, out_node, N, s);

  ln_kernel<<<(M + 7) / 8, LNB, 0, s>>>(mO, lnm_t_g, lnm_t_b, tmpM, M);
  gemm_up_gelu(tmpM, mt_W1, mt_b1, hid, M, s);
  gemm_down(hid, mt_W2, mt_b2, mO, out_tri, M, s);
}